// SpatialNeighborhoodAttention_70360154243805
// MI455X (gfx1250) — compile-verified
//
#include <hip/hip_runtime.h>
#include <math.h>

typedef float v8f __attribute__((ext_vector_type(8)));
typedef float v2f __attribute__((ext_vector_type(2)));

#define KDIM 768
#define BM 128
#define BN 64
#define KC 32
#define LDA 36   // 32 + 4 pad: keeps b128 LDS stores aligned, conflict-free row stride

// ---------------------------------------------------------------------------
// GEMM: Out[m][n] = sum_k A[m][k] * W[n][k] + bias[n]
// fp32 WMMA (v_wmma_f32_16x16x4_f32), 128x64 block tile, 8 waves, wave = 16x64
// ---------------------------------------------------------------------------
__global__ __launch_bounds__(256)
void gemm_bias_kernel(const float* __restrict__ A, const float* __restrict__ W,
                      const float* __restrict__ bias, float* __restrict__ Out,
                      int Nout) {
  __shared__ float lA[BM * LDA];
  __shared__ float lW[BN * LDA];
  const int tid  = threadIdx.x;
  const int wave = tid >> 5;
  const int lane = tid & 31;
  const int lm   = lane & 15;   // row-within-fragment
  const int kh   = lane >> 4;   // K half select
  const int mBase = blockIdx.y * BM;
  const int nBase = blockIdx.x * BN;

  v8f acc[4] = {};

  for (int kt = 0; kt < KDIM; kt += KC) {
    // stage A tile: 128 x 32 floats (4 float4 per thread)
#pragma unroll
    for (int i = 0; i < 4; ++i) {
      int s = tid + i * 256;
      int row = s >> 3, c4 = (s & 7) << 2;
      float4 v = *(const float4*)(A + (size_t)(mBase + row) * KDIM + kt + c4);
      *(float4*)(&lA[row * LDA + c4]) = v;
    }
    // stage W tile: 64 x 32 floats (2 float4 per thread)
#pragma unroll
    for (int i = 0; i < 2; ++i) {
      int s = tid + i * 256;
      int row = s >> 3, c4 = (s & 7) << 2;
      float4 v = *(const float4*)(W + (size_t)(nBase + row) * KDIM + kt + c4);
      *(float4*)(&lW[row * LDA + c4]) = v;
    }
    // prefetch next K tile (global_prefetch_b8)
    if (kt + KC < KDIM) {
      __builtin_prefetch(A + (size_t)(mBase + (tid >> 1)) * KDIM + kt + KC, 0, 0);
      if (tid < 128)
        __builtin_prefetch(W + (size_t)(nBase + (tid >> 1)) * KDIM + kt + KC, 0, 0);
    }
    __syncthreads();

    const float* aRow = &lA[(wave * 16 + lm) * LDA + 2 * kh];
    const float* wRow = &lW[lm * LDA + 2 * kh];
#pragma unroll
    for (int ks = 0; ks < KC; ks += 4) {
      v2f af = *(const v2f*)(aRow + ks);   // A frag: (M=lm, K=ks+2*kh+{0,1})
#pragma unroll
      for (int nt = 0; nt < 4; ++nt) {
        v2f bf = *(const v2f*)(wRow + nt * 16 * LDA + ks);
        acc[nt] = __builtin_amdgcn_wmma_f32_16x16x4_f32(
            false, af, false, bf, (short)0, acc[nt], false, false);
      }
    }
    __syncthreads();
  }

  // epilogue: C/D layout -> row M = r + 8*kh, col N = lm
#pragma unroll
  for (int nt = 0; nt < 4; ++nt) {
    int n = nBase + nt * 16 + lm;
    float bv = bias[n];
#pragma unroll
    for (int r = 0; r < 8; ++r) {
      int m = mBase + wave * 16 + r + 8 * kh;
      Out[(size_t)m * Nout + n] = acc[nt][r] + bv;
    }
  }
}

// ---------------------------------------------------------------------------
// wave32 reductions
// ---------------------------------------------------------------------------
__device__ __forceinline__ float wave_sum(float v) {
#pragma unroll
  for (int off = 16; off > 0; off >>= 1) v += __shfl_xor(v, off, 32);
  return v;
}

// ---------------------------------------------------------------------------
// Per-(token, head) LayerNorm on q,k + axial RoPE. One wave per (pos, head).
// In-place on qkv buffer: layout pos*2304 + head*288 + {q:0, k:96, v:192}
// ---------------------------------------------------------------------------
__global__ __launch_bounds__(256)
void ln_rope_kernel(float* __restrict__ qkv,
                    const float* __restrict__ qg, const float* __restrict__ qb,
                    const float* __restrict__ kg, const float* __restrict__ kb) {
  const int wave = threadIdx.x >> 5, lane = threadIdx.x & 31;
  const int task = blockIdx.x * 8 + wave;          // 9216*8 tasks
  const int head = task & 7;
  const int pos  = task >> 3;
  const int y = (pos / 48) % 48;
  const int x = pos % 48;
  float* base = qkv + (size_t)pos * 2304 + head * 288;

  // axial freqs: base_i = pi * linspace(1,128,16)[i], repeated x2
  const float ph = -1.0f + 2.0f * (float)y / 47.0f;
  const float pw = -1.0f + 2.0f * (float)x / 47.0f;
  const int   fi = lane >> 1;
  const float bf = 3.14159265358979323846f * (1.0f + (float)fi * (127.0f / 15.0f));
  const float f0 = ph * bf;                // dims [0,32): h-axis
  const float f1 = pw * bf;                // dims [32,64): w-axis
  const float c0 = cosf(f0), s0 = sinf(f0);
  const float c1 = cosf(f1), s1 = sinf(f1);
  const float sgn = (lane & 1) ? 1.0f : -1.0f;   // rotate_half sign

#pragma unroll
  for (int t = 0; t < 2; ++t) {            // t=0: q, t=1: k
    float* p = base + t * 96;
    const float* g = t ? kg : qg;
    const float* b = t ? kb : qb;
    float v0 = p[lane], v1 = p[lane + 32], v2 = p[lane + 64];
    float mu = wave_sum(v0 + v1 + v2) * (1.0f / 96.0f);
    float d0 = v0 - mu, d1 = v1 - mu, d2 = v2 - mu;
    float var = wave_sum(d0 * d0 + d1 * d1 + d2 * d2) * (1.0f / 96.0f);
    float inv = rsqrtf(var + 1e-5f);
    v0 = d0 * inv * g[lane]      + b[lane];
    v1 = d1 * inv * g[lane + 32] + b[lane + 32];
    v2 = d2 * inv * g[lane + 64] + b[lane + 64];
    // RoPE on dims [0,64): pair (2i,2i+1) lives at lane^1, same slot
    float o0 = __shfl_xor(v0, 1, 32);
    float o1 = __shfl_xor(v1, 1, 32);
    v0 = v0 * c0 + sgn * s0 * o0;
    v1 = v1 * c1 + sgn * s1 * o1;
    p[lane] = v0; p[lane + 32] = v1; p[lane + 64] = v2;  // v2 pass-through
  }
}

// ---------------------------------------------------------------------------
// 3x3 neighborhood attention, one wave per (pos, head). d=96 -> 3 floats/lane.
// ---------------------------------------------------------------------------
__global__ __launch_bounds__(256)
void na2d_kernel(const float* __restrict__ qkv, float* __restrict__ outp) {
  const int wave = threadIdx.x >> 5, lane = threadIdx.x & 31;
  const int task = blockIdx.x * 8 + wave;
  const int head = task & 7;
  const int pos  = task >> 3;
  const int bb = pos / 2304;               // batch*time index
  const int y = (pos / 48) % 48;
  const int x = pos % 48;
  const int ry = min(max(y - 1, 0), 45);   // clipped 3x3 window start
  const int rx = min(max(x - 1, 0), 45);

  const float* qp = qkv + (size_t)pos * 2304 + head * 288;
  const float q0 = qp[lane], q1 = qp[lane + 32], q2 = qp[lane + 64];
  const float scale = 0.10206207261596575f;   // 96^-0.5

  float sc[9];
#pragma unroll
  for (int i = 0; i < 3; ++i)
#pragma unroll
    for (int j = 0; j < 3; ++j) {
      int np = (bb * 48 + ry + i) * 48 + (rx + j);
      const float* kp = qkv + (size_t)np * 2304 + head * 288 + 96;
      float d = q0 * kp[lane] + q1 * kp[lane + 32] + q2 * kp[lane + 64];
      sc[i * 3 + j] = wave_sum(d) * scale;
    }
  float m = sc[0];
#pragma unroll
  for (int n = 1; n < 9; ++n) m = fmaxf(m, sc[n]);
  float denom = 0.0f;
#pragma unroll
  for (int n = 0; n < 9; ++n) { sc[n] = expf(sc[n] - m); denom += sc[n]; }
  const float rinv = 1.0f / denom;

  float a0 = 0.0f, a1 = 0.0f, a2 = 0.0f;
#pragma unroll
  for (int i = 0; i < 3; ++i)
#pragma unroll
    for (int j = 0; j < 3; ++j) {
      int np = (bb * 48 + ry + i) * 48 + (rx + j);
      const float* vp = qkv + (size_t)np * 2304 + head * 288 + 192;
      float w = sc[i * 3 + j] * rinv;
      a0 += w * vp[lane]; a1 += w * vp[lane + 32]; a2 += w * vp[lane + 64];
    }
  float* op = outp + (size_t)pos * 768 + head * 96;
  op[lane] = a0; op[lane + 32] = a1; op[lane + 64] = a2;
}

// ---------------------------------------------------------------------------
extern "C" void kernel_launch(void* const* d_in, const int* in_sizes, int n_in,
                              void* d_out, int out_size, void* d_ws, size_t ws_size,
                              hipStream_t stream) {
  const float* x     = (const float*)d_in[0];
  const float* w_qkv = (const float*)d_in[1];
  const float* b_qkv = (const float*)d_in[2];
  const float* w_out = (const float*)d_in[3];
  const float* b_out = (const float*)d_in[4];
  const float* qg    = (const float*)d_in[5];
  const float* qb    = (const float*)d_in[6];
  const float* kg    = (const float*)d_in[7];
  const float* kb    = (const float*)d_in[8];
  float* outp = (float*)d_out;

  float* qkv  = (float*)d_ws;                       // 9216 x 2304 f32 (~85 MB)
  float* attn = qkv + (size_t)9216 * 2304;          // 9216 x 768  f32 (~28 MB)

  // 1) QKV projection: M=9216, N=2304, K=768
  gemm_bias_kernel<<<dim3(2304 / BN, 9216 / BM), 256, 0, stream>>>(
      x, w_qkv, b_qkv, qkv, 2304);
  // 2) LayerNorm(q,k) + axial RoPE, in place
  ln_rope_kernel<<<dim3(9216), 256, 0, stream>>>(qkv, qg, qb, kg, kb);
  // 3) 3x3 neighborhood attention -> attn (9216 x 768)
  na2d_kernel<<<dim3(9216), 256, 0, stream>>>(qkv, attn);
  // 4) output projection: M=9216, N=768, K=768
  gemm_bias_kernel<<<dim3(768 / BN, 9216 / BM), 256, 0, stream>>>(
      attn, w_out, b_out, outp, 768);
}